// U_DCI_77378130805249
// MI455X (gfx1250) — compile-verified
//
#include <hip/hip_runtime.h>
#include <math.h>

#define IN_DIM 64
#define HID    128
#define NCLUS  16

typedef __attribute__((ext_vector_type(2))) float v2f;
typedef __attribute__((ext_vector_type(8))) float v8f;

__device__ __forceinline__ void gAtomicAdd(float* p, float v) {
  __hip_atomic_fetch_add(p, v, __ATOMIC_RELAXED, __HIP_MEMORY_SCOPE_AGENT);
}
__device__ __forceinline__ void sAtomicAdd(float* p, float v) {
  __hip_atomic_fetch_add(p, v, __ATOMIC_RELAXED, __HIP_MEMORY_SCOPE_WORKGROUP);
}
__device__ __forceinline__ float softplusf(float x) {
  // numerically stable softplus
  return fmaxf(x, 0.0f) + log1pf(expf(-fabsf(x)));
}

// ---------------------------------------------------------------------------
// 1) Edge scatter-sum: agg[dst] += x[src], one thread per (edge, dim)
// ---------------------------------------------------------------------------
__global__ __launch_bounds__(256) void k_aggregate(const float* __restrict__ x,
                                                   const int* __restrict__ src,
                                                   const int* __restrict__ dst,
                                                   float* __restrict__ agg,
                                                   int nEdges) {
  int t = blockIdx.x * 256 + threadIdx.x;
  int e = t >> 6;        // edge
  int d = t & 63;        // dim 0..63
  if (e < nEdges) {
    int s  = src[e];
    int dd = dst[e];
    float v = x[(long long)s * IN_DIM + d];
    gAtomicAdd(&agg[(long long)dd * IN_DIM + d], v);
  }
}

// ---------------------------------------------------------------------------
// 2) Fused GIN MLP: Z = relu((x+agg)@W1 + b1) @ W2 + b2, via f32 WMMA
//    One block = 16 output rows; 8 waves each own a 16-column tile.
// ---------------------------------------------------------------------------
__global__ __launch_bounds__(256) void k_encoder(const float* __restrict__ x,
                                                 const float* __restrict__ agg,
                                                 const float* __restrict__ W1,
                                                 const float* __restrict__ b1,
                                                 const float* __restrict__ W2,
                                                 const float* __restrict__ b2,
                                                 float* __restrict__ Z,
                                                 int N) {
  __shared__ float Xs[16][IN_DIM + 4];  // x+agg tile, padded
  __shared__ float Hs[16][HID + 4];     // hidden tile, padded (stride 132 kills bank conflicts)

  const int m0  = blockIdx.x * 16;
  const int tid = threadIdx.x;

  // Phase 0: stage Xs = x + agg for 16 rows (zero-pad past N)
  for (int i = tid; i < 16 * IN_DIM; i += 256) {
    int r = i >> 6, col = i & 63;
    long long g = (long long)(m0 + r) * IN_DIM + col;
    Xs[r][col] = (m0 + r < N) ? (x[g] + agg[g]) : 0.0f;
  }
  __syncthreads();

  const int lane  = tid & 31;
  const int w     = tid >> 5;     // wave id -> column tile
  const int n0    = w * 16;
  const int mrow  = lane & 15;    // M index of A rows / N index of B,C cols
  const int khalf = lane >> 4;    // 0: K pair {0,1}; 1: K pair {2,3}

  // Phase 1: H = relu(Xs @ W1 + b1), K = 64 (16 WMMA steps)
  v8f acc;
  {
    float bias = b1[n0 + mrow];
    for (int r = 0; r < 8; ++r) acc[r] = bias;
  }
  for (int s = 0; s < 16; ++s) {
    int k = s * 4 + khalf * 2;
    v2f a, b;
    a.x = Xs[mrow][k];
    a.y = Xs[mrow][k + 1];
    b.x = W1[(long long)k * HID + n0 + mrow];
    b.y = W1[(long long)(k + 1) * HID + n0 + mrow];
    acc = __builtin_amdgcn_wmma_f32_16x16x4_f32(false, a, false, b,
                                                (short)0, acc, false, false);
  }
  for (int r = 0; r < 8; ++r) {
    float v = acc[r] > 0.0f ? acc[r] : 0.0f;       // relu
    Hs[r + 8 * khalf][n0 + mrow] = v;              // C/D layout: M = r + 8*(lane/16)
  }
  __syncthreads();

  // Phase 2: Z = Hs @ W2 + b2, K = 128 (32 WMMA steps)
  {
    float bias = b2[n0 + mrow];
    for (int r = 0; r < 8; ++r) acc[r] = bias;
  }
  for (int s = 0; s < 32; ++s) {
    int k = s * 4 + khalf * 2;
    v2f a, b;
    a.x = Hs[mrow][k];
    a.y = Hs[mrow][k + 1];
    b.x = W2[(long long)k * HID + n0 + mrow];
    b.y = W2[(long long)(k + 1) * HID + n0 + mrow];
    acc = __builtin_amdgcn_wmma_f32_16x16x4_f32(false, a, false, b,
                                                (short)0, acc, false, false);
  }
  for (int r = 0; r < 8; ++r) {
    int row = m0 + r + 8 * khalf;
    if (row < N) Z[(long long)row * HID + n0 + mrow] = acc[r];
  }
}

// ---------------------------------------------------------------------------
// 3) Per-cluster sums of h1 + member counts (LDS-staged, then global atomics)
//    Block covers 256 rows; 256 threads = 2 rows x 128 dims at a time.
// ---------------------------------------------------------------------------
__global__ __launch_bounds__(256) void k_cluster_sum(const float* __restrict__ Z1,
                                                     const int* __restrict__ cid,
                                                     float* __restrict__ csum,
                                                     float* __restrict__ counts,
                                                     int N) {
  __shared__ float ls[NCLUS * HID];
  __shared__ float lc[NCLUS];
  for (int i = threadIdx.x; i < NCLUS * HID; i += 256) ls[i] = 0.0f;
  if (threadIdx.x < NCLUS) lc[threadIdx.x] = 0.0f;
  __syncthreads();

  const int r0   = blockIdx.x * 256;
  const int half = threadIdx.x >> 7;   // 0 or 1
  const int d    = threadIdx.x & 127;
  for (int rr = half; rr < 256; rr += 2) {
    int row = r0 + rr;
    if (row < N) {
      int k = cid[row];
      sAtomicAdd(&ls[k * HID + d], Z1[(long long)row * HID + d]);
      if (d == 0) sAtomicAdd(&lc[k], 1.0f);
    }
  }
  __syncthreads();

  for (int i = threadIdx.x; i < NCLUS * HID; i += 256) gAtomicAdd(&csum[i], ls[i]);
  if (threadIdx.x < NCLUS) gAtomicAdd(&counts[threadIdx.x], lc[threadIdx.x]);
}

// ---------------------------------------------------------------------------
// 4) c = sigmoid(csum/counts); u_k = Wd @ c_k  (single block)
// ---------------------------------------------------------------------------
__global__ __launch_bounds__(256) void k_summary(const float* __restrict__ csum,
                                                 const float* __restrict__ counts,
                                                 const float* __restrict__ Wd,
                                                 float* __restrict__ u) {
  __shared__ float cS[NCLUS * HID];
  for (int i = threadIdx.x; i < NCLUS * HID; i += 256) {
    float cnt = fmaxf(counts[i >> 7], 1.0f);
    float m   = csum[i] / cnt;
    cS[i] = 1.0f / (1.0f + expf(-m));
  }
  __syncthreads();
  for (int i = threadIdx.x; i < NCLUS * HID; i += 256) {
    int k = i >> 7, d = i & 127;
    float s = 0.0f;
    const float* ck = &cS[k * HID];
    for (int e = 0; e < HID; ++e) s += Wd[(long long)d * HID + e] * ck[e];
    u[k * HID + d] = s;
  }
}

// ---------------------------------------------------------------------------
// 5) Per-node bilinear scores + BCE, one wave32 per node, per-cluster atomics
// ---------------------------------------------------------------------------
__global__ __launch_bounds__(256) void k_score(const float* __restrict__ Z1,
                                               const float* __restrict__ Z2,
                                               const float* __restrict__ u,
                                               const int* __restrict__ cid,
                                               const float* __restrict__ bd,
                                               float* __restrict__ clSums,
                                               int N) {
  int node = (blockIdx.x * 256 + threadIdx.x) >> 5;
  int lane = threadIdx.x & 31;
  if (node >= N) return;   // wave-uniform exit

  int k = cid[node];
  const float* un = u + k * HID;
  long long base = (long long)node * HID;
  float s1 = 0.0f, s2 = 0.0f;
  for (int j = lane; j < HID; j += 32) {
    float uv = un[j];
    s1 += Z1[base + j] * uv;
    s2 += Z2[base + j] * uv;
  }
  for (int off = 16; off > 0; off >>= 1) {
    s1 += __shfl_down(s1, off, 32);
    s2 += __shfl_down(s2, off, 32);
  }
  if (lane == 0) {
    float b   = bd[0];
    float sc1 = s1 + b, sc2 = s2 + b;
    float pn  = softplusf(-sc1) + softplusf(sc2);
    gAtomicAdd(&clSums[k], pn);
  }
}

// ---------------------------------------------------------------------------
// 6) Final reduction: mean over clusters of clSums[k] / (2*counts[k])
// ---------------------------------------------------------------------------
__global__ void k_final(const float* __restrict__ clSums,
                        const float* __restrict__ counts,
                        float* __restrict__ out) {
  int t = threadIdx.x;
  float v = 0.0f;
  if (t < NCLUS) v = clSums[t] / (2.0f * fmaxf(counts[t], 1.0f));
  for (int off = 16; off > 0; off >>= 1) v += __shfl_down(v, off, 32);
  if (t == 0) out[0] = v / (float)NCLUS;
}

// ---------------------------------------------------------------------------
extern "C" void kernel_launch(void* const* d_in, const int* in_sizes, int n_in,
                              void* d_out, int out_size, void* d_ws, size_t ws_size,
                              hipStream_t stream) {
  const float* feats = (const float*)d_in[0];
  const float* shuf  = (const float*)d_in[1];
  const float* W1    = (const float*)d_in[2];
  const float* b1    = (const float*)d_in[3];
  const float* W2    = (const float*)d_in[4];
  const float* b2    = (const float*)d_in[5];
  const float* Wd    = (const float*)d_in[6];
  const float* bd    = (const float*)d_in[7];
  const int*   edges = (const int*)d_in[8];   // [2, E] (int32 under default JAX)
  const int*   cid   = (const int*)d_in[9];

  const int N = in_sizes[0] / IN_DIM;   // 50000
  const int E = in_sizes[8] / 2;        // 800000
  const int* src = edges;
  const int* dst = edges + E;

  // Workspace layout (floats)
  float* ws     = (float*)d_ws;
  float* agg    = ws;                                  // N*64 (reused for both passes)
  float* Z1     = agg + (size_t)N * IN_DIM;            // N*128
  float* Z2     = Z1  + (size_t)N * HID;               // N*128
  float* csum   = Z2  + (size_t)N * HID;               // 16*128
  float* counts = csum + NCLUS * HID;                  // 16
  float* u      = counts + NCLUS;                      // 16*128
  float* clSums = u + NCLUS * HID;                     // 16

  const int aggBlocks = (E * 64 + 255) / 256;
  const int encBlocks = (N + 15) / 16;
  const int csBlocks  = (N + 255) / 256;
  const int scBlocks  = ((size_t)N * 32 + 255) / 256;

  // ---- pass 1: positive embeddings ----
  hipMemsetAsync(agg, 0, (size_t)N * IN_DIM * sizeof(float), stream);
  k_aggregate<<<aggBlocks, 256, 0, stream>>>(feats, src, dst, agg, E);
  k_encoder<<<encBlocks, 256, 0, stream>>>(feats, agg, W1, b1, W2, b2, Z1, N);

  // ---- pass 2: negative (shuffled) embeddings ----
  hipMemsetAsync(agg, 0, (size_t)N * IN_DIM * sizeof(float), stream);
  k_aggregate<<<aggBlocks, 256, 0, stream>>>(shuf, src, dst, agg, E);
  k_encoder<<<encBlocks, 256, 0, stream>>>(shuf, agg, W1, b1, W2, b2, Z2, N);

  // ---- cluster summaries + discriminator vectors ----
  hipMemsetAsync(csum, 0, (2 * NCLUS * HID + 2 * NCLUS) * sizeof(float), stream);
  k_cluster_sum<<<csBlocks, 256, 0, stream>>>(Z1, cid, csum, counts, N);
  k_summary<<<1, 256, 0, stream>>>(csum, counts, Wd, u);

  // ---- scores + loss ----
  k_score<<<scBlocks, 256, 0, stream>>>(Z1, Z2, u, cid, bd, clSums, N);
  k_final<<<1, 32, 0, stream>>>(clSums, counts, (float*)d_out);
}